// MoE_64098091925598
// MI455X (gfx1250) — compile-verified
//
#include <hip/hip_runtime.h>
#include <hip/hip_bf16.h>

// ---------------------------------------------------------------------------
// MoE LoRA + base linear, fp32, CDNA5 (gfx1250).
//   WMMA f32 16x16x4 compute + Tensor Data Mover (TDM) double-buffered staging.
//   N=8192 tokens, D=1024, E=8 experts, R=16 rank, top-2 gating.
// ---------------------------------------------------------------------------

#define N_TOK 8192
#define D_MOD 1024
#define E_EXP 8
#define R_RANK 16
#define RC 128            // E*R = LoRA inner columns
#define LSTR 36           // LDS row stride (floats): 32 data + 4 pad -> bank-conflict-free
#define TILE_FLOATS (128 * LSTR)
#define TILE_BYTES  (TILE_FLOATS * 4)

typedef __attribute__((ext_vector_type(2))) float v2f;
typedef __attribute__((ext_vector_type(8))) float v8f;
typedef __attribute__((ext_vector_type(4))) unsigned int u32x4;
typedef __attribute__((ext_vector_type(8))) int i32x8;
typedef __attribute__((ext_vector_type(4))) int i32x4;

// ---------------------------------------------------------------------------
// TDM: DMA a 2D tile [tile_dim1=128 rows] x [tile_dim0=32 DWORDs] from global
// (row stride = strideElems f32) into LDS at lds_byte_off, padding each row
// with 4 DWORDs -> LDS layout row*36 + k, matching the WMMA fragment reads.
// D# packing per CDNA5 ISA 8.3/8.4 (group0 128b, group1 256b; groups 2/3 = 0).
// This toolchain exposes the 6-arg builtin: (g0, g1, g2, g3, i32x8, cpol).
// ---------------------------------------------------------------------------
__device__ __forceinline__ void tdm_load_tile_2d(unsigned lds_byte_off,
                                                 const float* gptr,
                                                 unsigned strideElems) {
  unsigned long long ga = (unsigned long long)(size_t)(const void*)gptr;
  u32x4 g0;
  g0.x = 1u;                                            // count=1 (valid), flags 0
  g0.y = lds_byte_off;                                  // lds_addr
  g0.z = (unsigned)ga;                                  // global_addr[31:0]
  g0.w = (unsigned)((ga >> 32) & 0x1FFFFFFull) | (2u << 30); // addr[56:32] | type=2
  i32x8 g1;
  // data_size=2 (4B) @16 | pad_enable @20 | pad_interval=4 (32 DWORDs) @22
  // | pad_amount=3 (4 DWORDs) @25
  g1[0] = (int)((2u << 16) | (1u << 20) | (4u << 22) | (3u << 25));
  g1[1] = (int)((1024u & 0xFFFFu) << 16);               // tensor_dim0 = 1024 (lo16)
  g1[2] = (int)(((1024u >> 16) & 0xFFFFu) |             // tensor_dim0 hi16
                ((1048576u & 0xFFFFu) << 16));          // tensor_dim1 = 1<<20 (lo16)
  g1[3] = (int)(((1048576u >> 16) & 0xFFFFu) |          // tensor_dim1 hi16
                (32u << 16));                           // tile_dim0 = 32 elems
  g1[4] = (int)(128u);                                  // tile_dim1 = 128, tile_dim2 = 0
  g1[5] = (int)strideElems;                             // tensor_dim0_stride[31:0]
  g1[6] = 0;                                            // stride hi | dim1_stride lo
  g1[7] = 0;
  i32x4 z4 = {0, 0, 0, 0};
  i32x8 z8 = {0, 0, 0, 0, 0, 0, 0, 0};
  __builtin_amdgcn_tensor_load_to_lds(g0, g1, z4, z4, z8, 0);
}

// ---------------------------------------------------------------------------
// Kernel 1: gating. One wave32 per token. logits = x @ Wg^T, top-2, softmax,
// scatter into dense combine[N, E].
// ---------------------------------------------------------------------------
__global__ __launch_bounds__(256) void gate_kernel(
    const float* __restrict__ x, const float* __restrict__ Wg,
    float* __restrict__ combine) {
  int tok  = (blockIdx.x * blockDim.x + threadIdx.x) >> 5;
  int lane = threadIdx.x & 31;
  if (tok >= N_TOK) return;

  const float4* xr = (const float4*)(x + (size_t)tok * D_MOD);
  float acc[E_EXP];
#pragma unroll
  for (int e = 0; e < E_EXP; ++e) acc[e] = 0.f;

#pragma unroll
  for (int i = 0; i < D_MOD / 4 / 32; ++i) {
    int idx = lane + i * 32;
    float4 xv = xr[idx];
#pragma unroll
    for (int e = 0; e < E_EXP; ++e) {
      float4 wv = ((const float4*)(Wg + (size_t)e * D_MOD))[idx];
      acc[e] += xv.x * wv.x + xv.y * wv.y + xv.z * wv.z + xv.w * wv.w;
    }
  }
#pragma unroll
  for (int e = 0; e < E_EXP; ++e) {
    float v = acc[e];
#pragma unroll
    for (int off = 16; off > 0; off >>= 1) v += __shfl_xor(v, off, 32);
    acc[e] = v;
  }
  if (lane == 0) {
    int i0 = 0; float v0 = acc[0];
#pragma unroll
    for (int e = 1; e < E_EXP; ++e) if (acc[e] > v0) { v0 = acc[e]; i0 = e; }
    int i1 = -1; float v1 = -3.4e38f;
#pragma unroll
    for (int e = 0; e < E_EXP; ++e) if (e != i0 && acc[e] > v1) { v1 = acc[e]; i1 = e; }
    float e0 = __expf(v0 - v0), e1 = __expf(v1 - v0);
    float inv = 1.f / (e0 + e1);
    float w0 = e0 * inv, w1 = e1 * inv;
#pragma unroll
    for (int e = 0; e < E_EXP; ++e)
      combine[(size_t)tok * E_EXP + e] = (e == i0) ? w0 : ((e == i1) ? w1 : 0.f);
  }
}

// ---------------------------------------------------------------------------
// WMMA inner tile: wave computes 16 rows x 128 cols (8 accumulator tiles).
// A-frag (16x4 f32): lane%16 = M, lane/16 picks K pair {2hi,2hi+1}  -> b64 read
// B-frag (4x16 f32): lane%16 = N, lane/16 picks K pair               -> b64 read
// ---------------------------------------------------------------------------
__device__ __forceinline__ void mma_tile(const float* __restrict__ ldsA,
                                         const float* __restrict__ ldsB,
                                         int wid, int m, int hi, v8f acc[8]) {
  const float* arow = ldsA + (wid * 16 + m) * LSTR + 2 * hi;
  v2f afrag[8];
#pragma unroll
  for (int ks = 0; ks < 8; ++ks) afrag[ks] = *(const v2f*)(arow + 4 * ks);
#pragma unroll
  for (int ct = 0; ct < 8; ++ct) {
    const float* brow = ldsB + (ct * 16 + m) * LSTR + 2 * hi;
#pragma unroll
    for (int ks = 0; ks < 8; ++ks) {
      v2f bf = *(const v2f*)(brow + 4 * ks);
      acc[ct] = __builtin_amdgcn_wmma_f32_16x16x4_f32(
          false, afrag[ks], false, bf, (short)0, acc[ct], false, false);
    }
  }
}

// ---------------------------------------------------------------------------
// Kernel 2: g[n, e*16+r] = combine[n,e] * (x[n,:] @ A[e,:,r]).
// One WG per 128 token rows; all 128 (e,r) columns. K=1024 staged in 32-slabs.
// ---------------------------------------------------------------------------
__global__ __launch_bounds__(256) void lora_h_kernel(
    const float* __restrict__ x, const float* __restrict__ Aw,
    const float* __restrict__ combine, float* __restrict__ g) {
  __shared__ float ldsA[TILE_FLOATS];
  __shared__ float ldsB[TILE_FLOATS];
  const int tid = threadIdx.x;
  const int wid = tid >> 5, lane = tid & 31;
  const int m = lane & 15, hi = lane >> 4;
  const int nbase = blockIdx.x * 128;

  v8f acc[8] = {};

  for (int k0 = 0; k0 < D_MOD; k0 += 32) {
    {
      int r0 = tid >> 3, kq = (tid & 7) * 4;
#pragma unroll
      for (int p = 0; p < 4; ++p) {
        int r = p * 32 + r0;
        float4 v = *(const float4*)(x + (size_t)(nbase + r) * D_MOD + k0 + kq);
        float* dst = &ldsA[r * LSTR + kq];
        dst[0] = v.x; dst[1] = v.y; dst[2] = v.z; dst[3] = v.w;
      }
    }
    {
      int e = tid >> 5, kk0 = (tid >> 2) & 7, r4 = (tid & 3) * 4;
#pragma unroll
      for (int p = 0; p < 4; ++p) {
        int kk = p * 8 + kk0;
        float4 v = *(const float4*)(Aw + (size_t)e * D_MOD * R_RANK +
                                    (size_t)(k0 + kk) * R_RANK + r4);
        int c = e * 16 + r4;
        ldsB[(c + 0) * LSTR + kk] = v.x;
        ldsB[(c + 1) * LSTR + kk] = v.y;
        ldsB[(c + 2) * LSTR + kk] = v.z;
        ldsB[(c + 3) * LSTR + kk] = v.w;
      }
    }
    __syncthreads();
    mma_tile(ldsA, ldsB, wid, m, hi, acc);
    __syncthreads();
  }

  const int rowb = nbase + wid * 16;
#pragma unroll
  for (int e = 0; e < 8; ++e) {
#pragma unroll
    for (int i = 0; i < 8; ++i) {
      int row = rowb + i + 8 * hi;
      float cmb = combine[(size_t)row * E_EXP + e];
      g[(size_t)row * RC + e * 16 + m] = acc[e][i] * cmb;
    }
  }
}

// ---------------------------------------------------------------------------
// Kernel 3: out = x @ W_base^T + g @ B_flat + b_base.
// 128x128 output tile. Phase 1 (K=1024): TDM double-buffered LDS staging —
// wave 0 issues tensor_load_to_lds for slab s+1 while all waves run WMMA on
// slab s (TENSORcnt pipelining). Phase 2 (K=128): manual staging.
// Dynamic LDS: 2 x Atile + 2 x Btile = 4 * 128*36 floats = 72 KB.
// ---------------------------------------------------------------------------
__global__ __launch_bounds__(256) void out_kernel(
    const float* __restrict__ x, const float* __restrict__ Bw,
    const float* __restrict__ Wb, const float* __restrict__ bb,
    const float* __restrict__ g, float* __restrict__ out) {
  extern __shared__ float smem[];
  float* const A0 = smem;
  float* const A1 = smem + TILE_FLOATS;
  float* const B0 = smem + 2 * TILE_FLOATS;
  float* const B1 = smem + 3 * TILE_FLOATS;

  const int tid = threadIdx.x;
  const int wid = tid >> 5, lane = tid & 31;
  const int m = lane & 15, hi = lane >> 4;
  const int nbase = blockIdx.x * 128;
  const int cbase = blockIdx.y * 128;

  v8f acc[8];
#pragma unroll
  for (int ct = 0; ct < 8; ++ct) {
    float b = bb[cbase + ct * 16 + m];
#pragma unroll
    for (int i = 0; i < 8; ++i) acc[ct][i] = b;
  }

  // LDS byte offset of dynamic smem (runtime scalar; flat-LDS low 32 bits).
  const unsigned sOff = (unsigned)(unsigned long long)(size_t)(void*)smem;

  // ---- phase 1: base GEMM, K over D=1024, TDM double-buffered ----
  if (tid < 32) {                                   // wave 0 only (uniform branch)
    tdm_load_tile_2d(sOff,                 x  + (size_t)nbase * D_MOD, D_MOD);
    tdm_load_tile_2d(sOff + 2u * TILE_BYTES, Wb + (size_t)cbase * D_MOD, D_MOD);
  }
  for (int s = 0; s < D_MOD / 32; ++s) {
    const int cur = s & 1;
    if (tid < 32) {
      if (s + 1 < D_MOD / 32) {
        const int k1 = (s + 1) * 32;
        const unsigned nb = (unsigned)(cur ^ 1);
        tdm_load_tile_2d(sOff + nb * TILE_BYTES,
                         x + (size_t)nbase * D_MOD + k1, D_MOD);
        tdm_load_tile_2d(sOff + (2u + nb) * TILE_BYTES,
                         Wb + (size_t)cbase * D_MOD + k1, D_MOD);
        __builtin_amdgcn_s_wait_tensorcnt(2);       // slab s done; s+1 in flight
      } else {
        __builtin_amdgcn_s_wait_tensorcnt(0);
      }
    }
    __syncthreads();                                // slab s visible to all waves
    mma_tile(cur ? A1 : A0, cur ? B1 : B0, wid, m, hi, acc);
    __syncthreads();                                // WAR: done reading buf[cur]
  }

  // ---- phase 2: LoRA up-proj, K over RC=128, manual staging ----
  float* const ldsA = A0;
  float* const ldsB = B0;
  for (int k0 = 0; k0 < RC; k0 += 32) {
    {
      int r0 = tid >> 3, kq = (tid & 7) * 4;
#pragma unroll
      for (int p = 0; p < 4; ++p) {
        int r = p * 32 + r0;
        float4 v = *(const float4*)(g + (size_t)(nbase + r) * RC + k0 + kq);
        float* dst = &ldsA[r * LSTR + kq];
        dst[0] = v.x; dst[1] = v.y; dst[2] = v.z; dst[3] = v.w;
      }
    }
    {
      int kk0 = tid >> 5, cq = (tid & 31) * 4;
#pragma unroll
      for (int p = 0; p < 4; ++p) {
        int kk = p * 8 + kk0;
        float4 v = *(const float4*)(Bw + (size_t)(k0 + kk) * D_MOD + cbase + cq);
        ldsB[(cq + 0) * LSTR + kk] = v.x;
        ldsB[(cq + 1) * LSTR + kk] = v.y;
        ldsB[(cq + 2) * LSTR + kk] = v.z;
        ldsB[(cq + 3) * LSTR + kk] = v.w;
      }
    }
    __syncthreads();
    mma_tile(ldsA, ldsB, wid, m, hi, acc);
    __syncthreads();
  }

  // ---- store C: element [i + 8*hi, m] of each 16x16 tile ----
  const int rowb = nbase + wid * 16;
#pragma unroll
  for (int ct = 0; ct < 8; ++ct) {
#pragma unroll
    for (int i = 0; i < 8; ++i) {
      int row = rowb + i + 8 * hi;
      out[(size_t)row * D_MOD + cbase + ct * 16 + m] = acc[ct][i];
    }
  }
}

// ---------------------------------------------------------------------------
extern "C" void kernel_launch(void* const* d_in, const int* in_sizes, int n_in,
                              void* d_out, int out_size, void* d_ws, size_t ws_size,
                              hipStream_t stream) {
  const float* x  = (const float*)d_in[0];
  const float* Wg = (const float*)d_in[1];
  const float* Aw = (const float*)d_in[2];
  const float* Bw = (const float*)d_in[3];
  const float* Wb = (const float*)d_in[4];
  const float* bb = (const float*)d_in[5];
  float* out = (float*)d_out;

  float* combine = (float*)d_ws;                       // [N, E]
  float* g = combine + (size_t)N_TOK * E_EXP;          // [N, RC]

  const size_t smem_bytes = (size_t)4 * TILE_BYTES;    // 72 KB dynamic LDS

  gate_kernel<<<N_TOK / 8, 256, 0, stream>>>(x, Wg, combine);
  lora_h_kernel<<<N_TOK / 128, 256, 0, stream>>>(x, Aw, combine, g);
  out_kernel<<<dim3(N_TOK / 128, D_MOD / 128), 256, smem_bytes, stream>>>(
      x, Bw, Wb, bb, g, out);
}